// ModelNew_3556232921926
// MI455X (gfx1250) — compile-verified
//
#include <hip/hip_runtime.h>
#include <hip/hip_bf16.h>
#include <math.h>

typedef __attribute__((ext_vector_type(2))) float v2f;
typedef __attribute__((ext_vector_type(8))) float v8f;

#define WSTRIDE 194                   // padded LDS row stride (floats); even -> aligned b64
#define PHW     34                    // 32 + 2 halo
#define PPLANE  (PHW*PHW)             // 1156 pixels per (b)
#define P_ELEMS (16*PPLANE*48)        // packed input [b][34][34][48ch], zero halo
#define WA_ELEMS (4*64*192)           // class weights [class][o][k]

// ---------------------------------------------------------------------------
// Kernel 0: zero the packed-input buffer (establishes the zero halo).
// ---------------------------------------------------------------------------
__global__ __launch_bounds__(256) void zero_p(float4* __restrict__ p) {
    p[blockIdx.x * 256 + threadIdx.x] = float4{0.f, 0.f, 0.f, 0.f};   // 221,952 float4
}

// ---------------------------------------------------------------------------
// Kernel 1: depth-collapse input into halo-padded pixel-major layout.
// ch i: sum_d x ; ch 16+i: x[d=0] ; ch 32+i: x[d=15]
// ---------------------------------------------------------------------------
__global__ __launch_bounds__(256) void prep_input(const float* __restrict__ x,
                                                  float* __restrict__ P) {
    int t  = blockIdx.x * 256 + threadIdx.x;   // 262144 threads: (b, i, hw)
    int hw = t & 1023;
    int i  = (t >> 10) & 15;
    int b  = t >> 14;
    int h  = hw >> 5, w = hw & 31;
    const float* xp = x + ((size_t)(b * 16 + i)) * 16 * 1024 + hw;
    float s = 0.f, x0 = 0.f, x15 = 0.f;
#pragma unroll
    for (int d = 0; d < 16; ++d) {
        float v = xp[(size_t)d * 1024];
        s += v;
        if (d == 0)  x0  = v;
        if (d == 15) x15 = v;
    }
    size_t base = ((size_t)b * PPLANE + (h + 1) * PHW + (w + 1)) * 48;
    P[base + i]      = s;
    P[base + 16 + i] = x0;
    P[base + 32 + i] = x15;
}

// ---------------------------------------------------------------------------
// Kernel 2: per-parity-class effective weights WA[c][o][k], k = tap*48 + ic,
// plus fused bias FB[o] = conv_bias[o] + bias[o].
// class c = e*2+ew; kh = (1-e)+2*th, kw = (1-ew)+2*tw.
// ic<16: sum_kd W ; 16..31: -W[kd=0] ; 32..47: -W[kd=3]  (depth-boundary fix)
// ---------------------------------------------------------------------------
__global__ __launch_bounds__(256) void prep_weights(const float* __restrict__ w,
                                                    const float* __restrict__ conv_bias,
                                                    const float* __restrict__ bias,
                                                    float* __restrict__ WA,
                                                    float* __restrict__ FB) {
    int t = blockIdx.x * 256 + threadIdx.x;    // 49152 threads
    int k = t % 192;
    int o = (t / 192) & 63;
    int c = t / (192 * 64);
    int e = c >> 1, ew = c & 1;
    int tap = k / 48, ic = k % 48;
    int kh = (1 - e)  + 2 * (tap >> 1);
    int kw = (1 - ew) + 2 * (tap & 1);
    float val;
    // weight flat layout: ((i*64 + o)*4 + kd)*16 + kh*4 + kw
    if (ic < 16) {
        int base = ((ic * 64 + o) * 4) * 16 + kh * 4 + kw;
        val = w[base] + w[base + 16] + w[base + 32] + w[base + 48];
    } else if (ic < 32) {
        val = -w[(((ic - 16) * 64 + o) * 4 + 0) * 16 + kh * 4 + kw];
    } else {
        val = -w[(((ic - 32) * 64 + o) * 4 + 3) * 16 + kh * 4 + kw];
    }
    WA[(size_t)(c * 64 + o) * 192 + k] = val;
    if (t < 64) FB[t] = conv_bias[t] + bias[t];
}

// ---------------------------------------------------------------------------
// Kernel 3: WMMA GEMM (M=64 out-ch, N=16 pixels/wave, K=192) + fused
// depth-mean scale, bias, channel softmax, tanh, *2.
// Block: 256 thr = 8 waves; fixed (b, e, ew, row-group g). Wave: (hsel, half).
// All B loads are unconditional (zero halo handles boundaries).
// ---------------------------------------------------------------------------
__global__ __launch_bounds__(256) void tconv_gemm(const float* __restrict__ P,
                                                  const float* __restrict__ WAg,
                                                  const float* __restrict__ FBg,
                                                  float* __restrict__ out) {
    __shared__ float WA_s[64 * WSTRIDE];
    __shared__ float FB_s[64];

    int tid  = threadIdx.x;
    int beta = blockIdx.x;                 // 512 blocks: b(4) e(1) ew(1) g(3)
    int g  = beta & 7;
    int ew = (beta >> 3) & 1;
    int e  = (beta >> 4) & 1;
    int b  = beta >> 5;
    int c  = e * 2 + ew;

    // stage this class's A matrix [64][192] into LDS (padded stride: no bank conflicts)
    const float* src = WAg + (size_t)c * (64 * 192);
    for (int idx = tid; idx < 64 * 192; idx += 256) {
        int row = idx / 192, col = idx - row * 192;
        WA_s[row * WSTRIDE + col] = src[idx];
    }
    if (tid < 64) FB_s[tid] = FBg[tid];
    __syncthreads();

    int wave = tid >> 5;
    int lane = tid & 31;
    int hsel = wave >> 1;                  // 0..3
    int half = wave & 1;                   // 0..1
    int n    = lane & 15;                  // N column (pixel) / A row
    int kg   = lane >> 4;                  // K pair select (frag layout)

    int ht    = 4 * g + hsel;              // 0..31
    int h_out = e + 2 * ht;
    int wt    = half * 16 + n;             // 0..31
    int w_out = ew + 2 * wt;

    v8f acc0 = {}, acc1 = {}, acc2 = {}, acc3 = {};
    const float* Pb = P + (size_t)b * ((size_t)PPLANE * 48);

    // per-tap base pointers into the halo-padded pixel-major input
    const float* bp[4];
#pragma unroll
    for (int tap = 0; tap < 4; ++tap) {
        int th = tap >> 1, tw = tap & 1;
        int hp = ht + e - th + 1;          // padded h coord, always in [0,33]
        int wp = wt + ew - tw + 1;         // padded w coord, always in [0,33]
        bp[tap] = Pb + ((size_t)hp * PHW + wp) * 48;
        __builtin_prefetch(bp[tap], 0, 1);
    }

#pragma unroll
    for (int tap = 0; tap < 4; ++tap) {
        const float* bpt = bp[tap];
        int kt = tap * 48;
#pragma unroll
        for (int j = 0; j < 12; ++j) {
            int kk = 4 * j + 2 * kg;                 // K offset within tap (even)
            v2f bf = *(const v2f*)&bpt[kk];          // two consecutive channels, one b64
            int kbase = kt + kk;
            v2f a0 = *(const v2f*)&WA_s[( 0 + n) * WSTRIDE + kbase];
            v2f a1 = *(const v2f*)&WA_s[(16 + n) * WSTRIDE + kbase];
            v2f a2 = *(const v2f*)&WA_s[(32 + n) * WSTRIDE + kbase];
            v2f a3 = *(const v2f*)&WA_s[(48 + n) * WSTRIDE + kbase];
            acc0 = __builtin_amdgcn_wmma_f32_16x16x4_f32(false, a0, false, bf, (short)0, acc0, false, false);
            acc1 = __builtin_amdgcn_wmma_f32_16x16x4_f32(false, a1, false, bf, (short)0, acc1, false, false);
            acc2 = __builtin_amdgcn_wmma_f32_16x16x4_f32(false, a2, false, bf, (short)0, acc2, false, false);
            acc3 = __builtin_amdgcn_wmma_f32_16x16x4_f32(false, a3, false, bf, (short)0, acc3, false, false);
        }
    }

    // ---- fused epilogue: depth-mean (/32) + bias, softmax over 64 ch, tanh*2
    // Lane n holds M rows kg*8+jj of each 16-row M tile; partner lane n^16 has
    // the other 8 rows -> one shfl_xor(16) completes each channel reduction.
    float vals[32];
#pragma unroll
    for (int jj = 0; jj < 8; ++jj) {
        vals[jj]      = acc0[jj] * 0.03125f + FB_s[ 0 + kg * 8 + jj];
        vals[8 + jj]  = acc1[jj] * 0.03125f + FB_s[16 + kg * 8 + jj];
        vals[16 + jj] = acc2[jj] * 0.03125f + FB_s[32 + kg * 8 + jj];
        vals[24 + jj] = acc3[jj] * 0.03125f + FB_s[48 + kg * 8 + jj];
    }
    float mx = vals[0];
#pragma unroll
    for (int q = 1; q < 32; ++q) mx = fmaxf(mx, vals[q]);
    mx = fmaxf(mx, __shfl_xor(mx, 16, 32));

    float ssum = 0.f;
#pragma unroll
    for (int q = 0; q < 32; ++q) { vals[q] = __expf(vals[q] - mx); ssum += vals[q]; }
    ssum += __shfl_xor(ssum, 16, 32);
    float rs = 1.0f / ssum;

    float* op = out + (size_t)(b * 64) * 4096 + h_out * 64 + w_out;
#pragma unroll
    for (int q = 0; q < 32; ++q) {
        int mt = q >> 3, jj = q & 7;
        int o  = mt * 16 + kg * 8 + jj;
        float p  = vals[q] * rs;               // softmax in (0,1)
        float t2 = __expf(2.0f * p);
        float r  = (t2 - 1.0f) / (t2 + 1.0f);  // tanh(p)
        op[(size_t)o * 4096] = r * 2.0f;       // * SCALE
    }
}

// ---------------------------------------------------------------------------
extern "C" void kernel_launch(void* const* d_in, const int* in_sizes, int n_in,
                              void* d_out, int out_size, void* d_ws, size_t ws_size,
                              hipStream_t stream) {
    const float* x  = (const float*)d_in[0];   // (16,16,16,32,32)
    const float* w  = (const float*)d_in[1];   // (16,64,4,4,4)
    const float* cb = (const float*)d_in[2];   // (64,)
    const float* bs = (const float*)d_in[3];   // (1,64,1,1,1)
    float* out = (float*)d_out;                // (16,64,1,64,64)

    float* P  = (float*)d_ws;                  // 3.55 MB halo-padded packed input
    float* WA = P + P_ELEMS;                   // 192 KB class weights
    float* FB = WA + WA_ELEMS;                 // 256 B fused bias

    zero_p      <<< 867, 256, 0, stream>>>((float4*)P);   // 221,952 float4 = P_ELEMS
    prep_input  <<<1024, 256, 0, stream>>>(x, P);
    prep_weights<<< 192, 256, 0, stream>>>(w, cb, bs, WA, FB);
    tconv_gemm  <<< 512, 256, 0, stream>>>(P, WA, FB, out);
}